// GCN_34359738415
// MI455X (gfx1250) — compile-verified
//
#include <hip/hip_runtime.h>

#define NN 10000
#define NE 640000
#define INF 128
#define HF  128
#define OF  64

typedef float v2f __attribute__((ext_vector_type(2)));
typedef float v8f __attribute__((ext_vector_type(8)));

// ---------------- elementwise helpers ----------------

__global__ void zero_f32(float* __restrict__ p, int n) {
    int i = blockIdx.x * blockDim.x + threadIdx.x;
    if (i < n) p[i] = 0.0f;
}

// one thread per edge: count in/out degrees
__global__ void degrees_kernel(const int* __restrict__ src, const int* __restrict__ dst,
                               float* __restrict__ deg_out, float* __restrict__ deg_in, int nE) {
    int e = blockIdx.x * blockDim.x + threadIdx.x;
    if (e < nE) {
        atomicAdd(&deg_out[src[e]], 1.0f);
        atomicAdd(&deg_in[dst[e]], 1.0f);
    }
}

__global__ void rsqrt_kernel(const float* __restrict__ deg_out, const float* __restrict__ deg_in,
                             float* __restrict__ rs_out, float* __restrict__ rs_in, int n) {
    int i = blockIdx.x * blockDim.x + threadIdx.x;
    if (i < n) {
        rs_out[i] = rsqrtf(fmaxf(deg_out[i], 1.0f));
        rs_in[i]  = rsqrtf(fmaxf(deg_in[i],  1.0f));
    }
}

// h[node, :] = x[node, :] * rs[node]   (float4 chunks, 128 feats = 32 chunks/row)
__global__ void scale_rows_kernel(const float* __restrict__ x, const float* __restrict__ rs,
                                  float* __restrict__ h, int nNodes) {
    int i = blockIdx.x * blockDim.x + threadIdx.x;   // chunk id
    int node = i >> 5;
    int c    = i & 31;
    if (node >= nNodes) return;
    float s = rs[node];
    float4 v = *(const float4*)(x + (size_t)node * 128 + c * 4);
    v.x *= s; v.y *= s; v.z *= s; v.w *= s;
    *(float4*)(h + (size_t)node * 128 + c * 4) = v;
}

// ---------------- SpMM scatter: one wave32 per edge ----------------
// lane l owns feats [4l, 4l+4): float4 gather from h[src], 4x f32 atomic add into agg[dst]
__global__ void scatter_kernel(const int* __restrict__ src, const int* __restrict__ dst,
                               const float* __restrict__ h, float* __restrict__ agg, int nE) {
    int gid  = blockIdx.x * blockDim.x + threadIdx.x;
    int e    = gid >> 5;
    int lane = gid & 31;
    if (e >= nE) return;
    int s = src[e];
    int d = dst[e];
    float4 v = *(const float4*)(h + (size_t)s * 128 + lane * 4);
    float* out = agg + (size_t)d * 128 + lane * 4;
    unsafeAtomicAdd(out + 0, v.x);
    unsafeAtomicAdd(out + 1, v.y);
    unsafeAtomicAdd(out + 2, v.z);
    unsafeAtomicAdd(out + 3, v.w);
}

// ---------------- fused scale + GEMM + bias (+ReLU) via fp32 WMMA ----------------
// out[M,N] = maybe_relu( (A * rs[row]) @ W + bias )
// Compile-time K/N/RELU: the K-loop fully unrolls into a straight chain of
// K/4 v_wmma_f32_16x16x4_f32 ops with immediate-offset load clauses.
// One wave per 16x16 output tile; blockDim.x = (N/16)*32; grid.x = M/16.
template <int K, int N, bool RELU>
__global__ void gemm_wmma_f32(const float* __restrict__ A,   // M x K (unscaled agg)
                              const float* __restrict__ rs,  // per-row dst normalization
                              const float* __restrict__ W,   // K x N
                              const float* __restrict__ bias,// N
                              float* __restrict__ out) {     // M x N
    int lane  = threadIdx.x & 31;
    int wave  = threadIdx.x >> 5;
    int mBase = blockIdx.x * 16;
    int nBase = wave * 16;
    int half  = lane >> 4;      // 0: K pair {0,1} / rows M=r ; 1: K pair {2,3} / rows M=8+r
    int l16   = lane & 15;
    int mRow  = mBase + l16;    // A-fragment row owned by this lane
    int nCol  = nBase + l16;    // B/C/D column owned by this lane

    float rscale = rs[mRow];
    const float* Arow = A + (size_t)mRow * K + half * 2;  // lane's K-pair base
    const float* Wcol = W + (size_t)half * 2 * N + nCol;  // lane's K-pair base in W

    v8f c = {};
    #pragma unroll
    for (int k = 0; k < K; k += 4) {
        // A fragment: 16x4 fp32 (lanes 0-15 -> K=k,k+1 ; lanes 16-31 -> K=k+2,k+3)
        v2f a;
        a.x = Arow[k]     * rscale;
        a.y = Arow[k + 1] * rscale;
        // B fragment: 4x16 fp32 (mirrored K striping, N = lane%16)
        v2f b;
        b.x = Wcol[(size_t)k * N];
        b.y = Wcol[(size_t)(k + 1) * N];
        // 8 args: (neg_a, A, neg_b, B, c_mod, C, reuse_a, reuse_b)
        c = __builtin_amdgcn_wmma_f32_16x16x4_f32(false, a, false, b, (short)0, c,
                                                  false, false);
    }

    float bv = bias[nCol];
    // C/D layout: VGPR r -> M = mBase + r (lanes 0-15), M = mBase + 8 + r (lanes 16-31)
    #pragma unroll
    for (int r = 0; r < 8; ++r) {
        int row = mBase + r + half * 8;
        float v = c[r] + bv;
        if (RELU) v = fmaxf(v, 0.0f);
        out[(size_t)row * N + nCol] = v;
    }
}

// ---------------- launch ----------------

extern "C" void kernel_launch(void* const* d_in, const int* in_sizes, int n_in,
                              void* d_out, int out_size, void* d_ws, size_t ws_size,
                              hipStream_t stream) {
    const float* in_feat = (const float*)d_in[0];
    const int*   src     = (const int*)d_in[1];
    const int*   dst     = (const int*)d_in[2];
    const float* W1      = (const float*)d_in[3];
    const float* b1      = (const float*)d_in[4];
    const float* W2      = (const float*)d_in[5];
    const float* b2      = (const float*)d_in[6];
    float* out = (float*)d_out;

    // workspace layout (floats), ~15.6 MB total
    float* ws      = (float*)d_ws;
    float* deg_out = ws;                       // NN
    float* deg_in  = ws + NN;                  // NN (contiguous with deg_out)
    float* rs_out  = ws + 2 * NN;              // NN
    float* rs_in   = ws + 3 * NN;              // NN
    float* hbuf    = ws + 4 * NN;              // NN*128  (scatter source)
    float* agg     = hbuf + (size_t)NN * 128;  // NN*128  (scatter accumulator)
    float* hidden  = agg  + (size_t)NN * 128;  // NN*128  (layer-1 activations)

    const int B = 256;
    const size_t featElems = (size_t)NN * 128;

    // --- degrees + normalization factors (shared by both layers) ---
    zero_f32<<<(2 * NN + B - 1) / B, B, 0, stream>>>(deg_out, 2 * NN);
    degrees_kernel<<<(NE + B - 1) / B, B, 0, stream>>>(src, dst, deg_out, deg_in, NE);
    rsqrt_kernel<<<(NN + B - 1) / B, B, 0, stream>>>(deg_out, deg_in, rs_out, rs_in, NN);

    // --- layer 1 ---
    scale_rows_kernel<<<(NN * 32 + B - 1) / B, B, 0, stream>>>(in_feat, rs_out, hbuf, NN);
    zero_f32<<<(int)((featElems + B - 1) / B), B, 0, stream>>>(agg, (int)featElems);
    scatter_kernel<<<(NE * 32) / B, B, 0, stream>>>(src, dst, hbuf, agg, NE);
    // hidden = relu((agg * rs_in) @ W1 + b1) ; N=128 -> 8 waves/block
    gemm_wmma_f32<INF, HF, true><<<NN / 16, (HF / 16) * 32, 0, stream>>>(
        agg, rs_in, W1, b1, hidden);

    // --- layer 2 ---
    scale_rows_kernel<<<(NN * 32 + B - 1) / B, B, 0, stream>>>(hidden, rs_out, hbuf, NN);
    zero_f32<<<(int)((featElems + B - 1) / B), B, 0, stream>>>(agg, (int)featElems);
    scatter_kernel<<<(NE * 32) / B, B, 0, stream>>>(src, dst, hbuf, agg, NE);
    // out = (agg * rs_in) @ W2 + b2 ; N=64 -> 4 waves/block
    gemm_wmma_f32<HF, OF, false><<<NN / 16, (OF / 16) * 32, 0, stream>>>(
        agg, rs_in, W2, b2, out);
}